// TransitionDown_16999480557971
// MI455X (gfx1250) — compile-verified
//
#include <hip/hip_runtime.h>

#define N_PTS  32768
#define C_IN   64
#define C_OUT  128
#define M_SAMP 8192
#define K_NBR  16

typedef __attribute__((ext_vector_type(2))) float v2f;
typedef __attribute__((ext_vector_type(8))) float v8f;

// ---------------------------------------------------------------------------
// 1) Farthest point sampling: one 1024-thread workgroup, points cached in
//    VGPRs, per-round argmax via wave32 butterfly + 32-wave LDS reduce.
//    Also emits sub_pos and sub_batch.
// ---------------------------------------------------------------------------
__launch_bounds__(1024, 1)
__global__ void fps_kernel(const float* __restrict__ pos,
                           const int* __restrict__ batch,
                           float* __restrict__ sub_pos,
                           int* __restrict__ sub_batch) {
  __shared__ float s_lx, s_ly, s_lz;
  __shared__ float s_bd[32];
  __shared__ int   s_bi[32];
  __shared__ float s_bx[32], s_by[32], s_bz[32];
  __shared__ int   s_idx[M_SAMP];

  const int tid  = threadIdx.x;
  const int lane = tid & 31;
  const int wv   = tid >> 5;

  // Cache 32 points per thread in registers.
  float px[32], py[32], pz[32], mind[32];
#pragma unroll
  for (int j = 0; j < 32; ++j) {
    const int p = tid + j * 1024;
    px[j] = pos[p * 3 + 0];
    py[j] = pos[p * 3 + 1];
    pz[j] = pos[p * 3 + 2];
    mind[j] = __builtin_inff();
  }
  if (tid == 0) {
    s_lx = pos[0]; s_ly = pos[1]; s_lz = pos[2];
    s_idx[0] = 0;  // scan emits `last` before each update; starts at 0
  }
  __syncthreads();

  for (int it = 0; it < M_SAMP; ++it) {
    const float lx = s_lx, ly = s_ly, lz = s_lz;

    // Local min-update + argmax over this thread's 32 points.
    float bd = -1.0f;
    int   bi = 0x7fffffff;
    float bx = 0.f, by = 0.f, bz = 0.f;
#pragma unroll
    for (int j = 0; j < 32; ++j) {
      const float dx = px[j] - lx, dy = py[j] - ly, dz = pz[j] - lz;
      const float d = dx * dx + dy * dy + dz * dz;
      const float m = fminf(mind[j], d);
      mind[j] = m;
      if (m > bd) {  // strict > keeps lowest index among ties (j ascending)
        bd = m; bi = tid + j * 1024;
        bx = px[j]; by = py[j]; bz = pz[j];
      }
    }

    // Wave32 butterfly argmax, carrying coords of the candidate.
#pragma unroll
    for (int off = 16; off > 0; off >>= 1) {
      const float od = __shfl_xor(bd, off, 32);
      const int   oi = __shfl_xor(bi, off, 32);
      const float ox = __shfl_xor(bx, off, 32);
      const float oy = __shfl_xor(by, off, 32);
      const float oz = __shfl_xor(bz, off, 32);
      if (od > bd || (od == bd && oi < bi)) {
        bd = od; bi = oi; bx = ox; by = oy; bz = oz;
      }
    }
    if (lane == 0) {
      s_bd[wv] = bd; s_bi[wv] = bi;
      s_bx[wv] = bx; s_by[wv] = by; s_bz[wv] = bz;
    }
    __syncthreads();

    if (tid == 0) {
      float fd = s_bd[0]; int fi = s_bi[0];
      float fx = s_bx[0], fy = s_by[0], fz = s_bz[0];
#pragma unroll
      for (int w = 1; w < 32; ++w) {
        if (s_bd[w] > fd || (s_bd[w] == fd && s_bi[w] < fi)) {
          fd = s_bd[w]; fi = s_bi[w];
          fx = s_bx[w]; fy = s_by[w]; fz = s_bz[w];
        }
      }
      if (it + 1 < M_SAMP) s_idx[it + 1] = fi;
      s_lx = fx; s_ly = fy; s_lz = fz;
    }
    __syncthreads();
  }

  // Emit sub_pos / sub_batch.
  for (int s = tid; s < M_SAMP; s += 1024) {
    const int i = s_idx[s];
    sub_pos[s * 3 + 0] = pos[i * 3 + 0];
    sub_pos[s * 3 + 1] = pos[i * 3 + 1];
    sub_pos[s * 3 + 2] = pos[i * 3 + 2];
    sub_batch[s] = batch[i];
  }
}

// ---------------------------------------------------------------------------
// 2) h = relu(x @ W + b) using V_WMMA_F32_16X16X4_F32.
//    Each wave: one 16-row stripe x all 8 column tiles (8 v8f accumulators).
//    16 k-blocks * 8 tiles = 128 WMMA per wave.
// ---------------------------------------------------------------------------
__launch_bounds__(256, 1)
__global__ void mlp_wmma_kernel(const float* __restrict__ x,
                                const float* __restrict__ W,
                                const float* __restrict__ bias,
                                float* __restrict__ h) {
  const int lane = threadIdx.x & 31;
  const int wv   = threadIdx.x >> 5;
  const int r0   = (blockIdx.x * 8 + wv) * 16;  // stripe base row
  const int lm   = lane & 15;                   // M (for A) / N (for B,C,D)
  const int hh   = lane >> 4;                   // lane-half selector

  v8f acc[8] = {};

#pragma unroll
  for (int kb = 0; kb < 16; ++kb) {
    const int k0 = kb * 4 + 2 * hh;  // A layout: half-wave holds K pair {k0,k0+1}
    v2f a;
    const float* ap = x + (size_t)(r0 + lm) * C_IN + k0;
    a.x = ap[0];
    a.y = ap[1];
#pragma unroll
    for (int ct = 0; ct < 8; ++ct) {
      const int c0 = ct * 16;
      v2f b;
      b.x = W[(size_t)k0 * C_OUT + c0 + lm];
      b.y = W[(size_t)(k0 + 1) * C_OUT + c0 + lm];
      acc[ct] = __builtin_amdgcn_wmma_f32_16x16x4_f32(
          false, a, false, b, (short)0, acc[ct], false, false);
    }
  }

  // Epilogue: bias + ReLU, scatter per C/D VGPR layout (M = v + 8*hh, N = lm).
#pragma unroll
  for (int ct = 0; ct < 8; ++ct) {
    const int col = ct * 16 + lm;
    const float bv = bias[col];
#pragma unroll
    for (int v = 0; v < 8; ++v) {
      const int row = r0 + v + 8 * hh;
      const float val = acc[ct][v] + bv;
      h[(size_t)row * C_OUT + col] = fmaxf(val, 0.0f);
    }
  }
}

// ---------------------------------------------------------------------------
// 3) KNN: thread-per-query, point cloud staged through LDS in 1024-pt tiles,
//    register-resident sorted top-16 (strict < == top_k tie behavior).
// ---------------------------------------------------------------------------
__launch_bounds__(256, 1)
__global__ void knn_kernel(const float* __restrict__ pos,
                           const float* __restrict__ sub_pos,
                           int* __restrict__ nbr) {
  __shared__ float s_px[1024], s_py[1024], s_pz[1024];

  const int q = blockIdx.x * 256 + threadIdx.x;
  const float qx = sub_pos[q * 3 + 0];
  const float qy = sub_pos[q * 3 + 1];
  const float qz = sub_pos[q * 3 + 2];

  float bd[K_NBR];
  int   bi[K_NBR];
#pragma unroll
  for (int k = 0; k < K_NBR; ++k) { bd[k] = __builtin_inff(); bi[k] = -1; }

  for (int t0 = 0; t0 < N_PTS; t0 += 1024) {
    __syncthreads();
    for (int j = threadIdx.x; j < 1024; j += 256) {
      const int p = t0 + j;
      s_px[j] = pos[p * 3 + 0];
      s_py[j] = pos[p * 3 + 1];
      s_pz[j] = pos[p * 3 + 2];
    }
    __syncthreads();
#pragma unroll 4
    for (int j = 0; j < 1024; ++j) {
      const float dx = qx - s_px[j], dy = qy - s_py[j], dz = qz - s_pz[j];
      const float d = dx * dx + dy * dy + dz * dz;
      if (d < bd[K_NBR - 1]) {
        bd[K_NBR - 1] = d;
        bi[K_NBR - 1] = t0 + j;
#pragma unroll
        for (int k = K_NBR - 1; k > 0; --k) {
          if (bd[k] < bd[k - 1]) {
            const float td = bd[k]; bd[k] = bd[k - 1]; bd[k - 1] = td;
            const int   ti = bi[k]; bi[k] = bi[k - 1]; bi[k - 1] = ti;
          }
        }
      }
    }
  }
#pragma unroll
  for (int k = 0; k < K_NBR; ++k) nbr[q * K_NBR + k] = bi[k];
}

// ---------------------------------------------------------------------------
// 4) out[q][c] = max_k h[nbr[q][k]][c]
// ---------------------------------------------------------------------------
__launch_bounds__(128, 1)
__global__ void gather_max_kernel(const float* __restrict__ h,
                                  const int* __restrict__ nbr,
                                  float* __restrict__ out) {
  __shared__ int s_n[K_NBR];
  const int q = blockIdx.x;
  if (threadIdx.x < K_NBR) s_n[threadIdx.x] = nbr[q * K_NBR + threadIdx.x];
  __syncthreads();
  const int c = threadIdx.x;
  float v = -__builtin_inff();
#pragma unroll
  for (int k = 0; k < K_NBR; ++k) {
    v = fmaxf(v, h[(size_t)s_n[k] * C_OUT + c]);
  }
  out[(size_t)q * C_OUT + c] = v;
}

// ---------------------------------------------------------------------------
extern "C" void kernel_launch(void* const* d_in, const int* in_sizes, int n_in,
                              void* d_out, int out_size, void* d_ws, size_t ws_size,
                              hipStream_t stream) {
  const float* x     = (const float*)d_in[0];  // [32768,64]
  const float* pos   = (const float*)d_in[1];  // [32768,3]
  const int*   batch = (const int*)d_in[2];    // [32768]
  const float* W     = (const float*)d_in[3];  // [64,128]
  const float* b     = (const float*)d_in[4];  // [128]

  float* out       = (float*)d_out;                      // [8192,128]
  float* sub_pos   = out + (size_t)M_SAMP * C_OUT;       // [8192,3]
  int*   sub_batch = (int*)(sub_pos + (size_t)M_SAMP * 3);  // [8192]

  float* h   = (float*)d_ws;  // [32768,128] = 16 MB
  int*   nbr = (int*)((char*)d_ws + (size_t)N_PTS * C_OUT * sizeof(float));  // [8192,16]

  fps_kernel<<<dim3(1), dim3(1024), 0, stream>>>(pos, batch, sub_pos, sub_batch);
  mlp_wmma_kernel<<<dim3(N_PTS / (8 * 16)), dim3(256), 0, stream>>>(x, W, b, h);
  knn_kernel<<<dim3(M_SAMP / 256), dim3(256), 0, stream>>>(pos, sub_pos, nbr);
  gather_max_kernel<<<dim3(M_SAMP), dim3(128), 0, stream>>>(h, nbr, out);
}